// HyperDRQN_59339268161757
// MI455X (gfx1250) — compile-verified
//
#include <hip/hip_runtime.h>
#include <hip/hip_bf16.h>
#include <math.h>

// ---------------------------------------------------------------------------
// HyperDRQN forward for MI455X (gfx1250), all GEMMs via V_WMMA_F32_16X16X4_F32
// B=32, N=512, IN_DIM=128, HID=256, HEADS=4, HD=64, ACT=32, rows = B*N = 16384
// GEMM A-tiles staged to LDS via GLOBAL_LOAD_ASYNC_TO_LDS_B128 when available.
// ---------------------------------------------------------------------------

typedef float v2f __attribute__((ext_vector_type(2)));
typedef float v8f __attribute__((ext_vector_type(8)));
typedef int   vsi4 __attribute__((vector_size(16)));   // matches builtin V4i

#define ROWS   16384   // B*N
#define HIDDIM 256
#define NN     512
#define NB     32
#define NHEADS 4
#define HDD    64
#define ACTD   32

#define AS_GLOBAL __attribute__((address_space(1)))
#define AS_SHARED __attribute__((address_space(3)))

#if __has_builtin(__builtin_amdgcn_global_load_async_to_lds_b128) && \
    __has_builtin(__builtin_amdgcn_s_wait_asynccnt)
#define USE_ASYNC_LDS 1
#else
#define USE_ASYNC_LDS 0
#endif

__device__ __forceinline__ v8f wmma4(v2f a, v2f b, v8f c) {
  // D = A(16x4 f32) * B(4x16 f32) + C(16x16 f32)
  return __builtin_amdgcn_wmma_f32_16x16x4_f32(
      /*neg_a=*/false, a, /*neg_b=*/false, b,
      /*c_mod=*/(short)0, c, /*reuse_a=*/false, /*reuse_b=*/false);
}

// ---------------------------------------------------------------------------
// Generic GEMM: C[M x Nout] = A[M x K] * W[Nout x K]^T + bias[n] (+ addrow[n])
//               (+ Cin[m,n]).
// block=(32,4): block tile = 16 rows x 128 cols, wave tile = 16x32 (2 accs).
// A tile (16 x K) staged once per block into LDS (async DMA when available),
// shared by all 4 waves; each LDS A fragment feeds 2 WMMAs.
// grid.x = ceil(Nout/128), grid.y = M/16.
// ---------------------------------------------------------------------------
template <int K>
__global__ void gemm_wmma_kernel(const float* __restrict__ A,
                                 const float* __restrict__ W,
                                 const float* __restrict__ bias,
                                 const float* __restrict__ addrow,
                                 const float* __restrict__ Cin,
                                 float* __restrict__ C,
                                 int Nout) {
  __shared__ float sA[16 * K];
  const int lane = threadIdx.x;          // 0..31
  const int wave = threadIdx.y;          // 0..3
  const int tid  = wave * 32 + lane;     // 0..127
  const int m0   = blockIdx.y * 16;

  // ---- cooperative stage of the 16 x K A tile into LDS (coalesced b128) ----
  constexpr int VEC = 16 * K / 4;        // number of float4's
#pragma unroll
  for (int i = tid; i < VEC; i += 128) { // exact multiple: no divergence
    const int row = i / (K / 4);
    const int c4  = (i % (K / 4)) * 4;
    const float* gsrc = A + (size_t)(m0 + row) * K + c4;
    float*       ldst = &sA[row * K + c4];
#if USE_ASYNC_LDS
    __builtin_amdgcn_global_load_async_to_lds_b128(
        (AS_GLOBAL vsi4*)gsrc, (AS_SHARED vsi4*)ldst, 0, 0);
#else
    *(float4*)ldst = *(const float4*)gsrc;
#endif
  }
#if USE_ASYNC_LDS
  __builtin_amdgcn_s_wait_asynccnt(0);
#endif
  __syncthreads();

  const int n0 = blockIdx.x * 128 + wave * 32;
  if (n0 >= Nout) return;                // wave-uniform; barrier already passed

  const int half = lane >> 4;            // 0: K=k,k+1   1: K=k+2,k+3
  const int l    = lane & 15;

  const float* sAp = sA + l * K + half * 2;                 // LDS A row l
  const float* Wp0 = W + (size_t)(n0 + l) * K + half * 2;   // B col n0+l
  const float* Wp1 = W + (size_t)(n0 + 16 + l) * K + half * 2;

  v8f acc0 = {}, acc1 = {};
  for (int k = 0; k < K; k += 4) {
    v2f a  = *(const v2f*)(sAp + k);     // ds_load_b64
    v2f b0 = *(const v2f*)(Wp0 + k);
    v2f b1 = *(const v2f*)(Wp1 + k);
    acc0 = wmma4(a, b0, acc0);           // A fragment reused twice
    acc1 = wmma4(a, b1, acc1);
  }

  const int col0 = n0 + l, col1 = n0 + 16 + l;
  float bb0 = bias ? bias[col0] : 0.0f;
  float bb1 = bias ? bias[col1] : 0.0f;
  if (addrow) { bb0 += addrow[col0]; bb1 += addrow[col1]; }
#pragma unroll
  for (int r = 0; r < 8; ++r) {
    const int row = m0 + half * 8 + r;
    float v0 = acc0[r] + bb0;
    float v1 = acc1[r] + bb1;
    if (Cin) {
      v0 += Cin[(size_t)row * Nout + col0];
      v1 += Cin[(size_t)row * Nout + col1];
    }
    C[(size_t)row * Nout + col0] = v0;
    C[(size_t)row * Nout + col1] = v1;
  }
}

// ---------------------------------------------------------------------------
// LayerNorm + exact GELU, in place.  One 256-thread block per row.
// ---------------------------------------------------------------------------
__global__ void ln_gelu_kernel(float* __restrict__ h,
                               const float* __restrict__ g,
                               const float* __restrict__ b, int D) {
  __shared__ float sred[256];
  const int tid = threadIdx.x;
  float* p = h + (size_t)blockIdx.x * D;

  float s = 0.0f;
  for (int i = tid; i < D; i += 256) s += p[i];
  sred[tid] = s; __syncthreads();
  for (int off = 128; off > 0; off >>= 1) {
    if (tid < off) sred[tid] += sred[tid + off];
    __syncthreads();
  }
  const float mu = sred[0] / (float)D;
  __syncthreads();

  float v = 0.0f;
  for (int i = tid; i < D; i += 256) { float d = p[i] - mu; v += d * d; }
  sred[tid] = v; __syncthreads();
  for (int off = 128; off > 0; off >>= 1) {
    if (tid < off) sred[tid] += sred[tid + off];
    __syncthreads();
  }
  const float rstd = rsqrtf(sred[0] / (float)D + 1e-5f);

  for (int i = tid; i < D; i += 256) {
    float y = (p[i] - mu) * rstd * g[i] + b[i];
    p[i] = 0.5f * y * (1.0f + erff(y * 0.70710678118654752f));
  }
}

// ---------------------------------------------------------------------------
// GAT per-head score projections: hl[b,h,n], hr[b,h,n]
// ---------------------------------------------------------------------------
__global__ void gat_score_kernel(const float* __restrict__ Wh,
                                 const float* __restrict__ gat_a,
                                 float* __restrict__ hl,
                                 float* __restrict__ hr) {
  const int idx = blockIdx.x * blockDim.x + threadIdx.x;  // B*N*HEADS = 65536
  const int h  = idx & (NHEADS - 1);
  const int bn = idx >> 2;
  const float* w  = Wh + (size_t)bn * HIDDIM + h * HDD;
  const float* al = gat_a + h * (2 * HDD);
  float sl = 0.0f, sr = 0.0f;
#pragma unroll 4
  for (int d = 0; d < HDD; ++d) {
    float v = w[d];
    sl += v * al[d];
    sr += v * al[HDD + d];
  }
  const int b = bn >> 9, n = bn & (NN - 1);
  hl[((size_t)b * NHEADS + h) * NN + n] = sl;
  hr[((size_t)b * NHEADS + h) * NN + n] = sr;
}

// ---------------------------------------------------------------------------
// Fused graph attention: masked leaky-relu softmax (rank-1 scores) + P @ Wh.
// Block = (32,4) = 128 threads handles one (b, head, 16-row tile).
// P tile (16 x 512) staged in LDS, consumed by 4 waves of WMMA (K=512).
// grid = (N/16, HEADS, B)
// ---------------------------------------------------------------------------
__global__ void gat_attn_kernel(const float* __restrict__ Wh,
                                const float* __restrict__ hl,
                                const float* __restrict__ hr,
                                const int* __restrict__ adj,
                                float* __restrict__ out) {
  __shared__ float sP[16 * NN];      // 32 KB probability tile
  __shared__ float sred[16][8];
  __shared__ float srow[32];         // [0..15] row max, [16..31] 1/rowsum

  const int nt = blockIdx.x, h = blockIdx.y, b = blockIdx.z;
  const int tid = threadIdx.y * 32 + threadIdx.x;   // 0..127
  const int r   = tid >> 3;                         // row 0..15
  const int t8  = tid & 7;                          // 0..7
  const int n   = nt * 16 + r;

  const float  hlv   = hl[((size_t)b * NHEADS + h) * NN + n];
  const float* hrb   = hr + ((size_t)b * NHEADS + h) * NN;
  const int*   adjr  = adj + ((size_t)b * NN + n) * NN;

  // pass 1: masked row max of leaky_relu(hl[n] + hr[m])
  float mx = -INFINITY;
  for (int m = t8; m < NN; m += 8) {
    if (adjr[m] != 0) {
      float e = hlv + hrb[m];
      e = (e >= 0.0f) ? e : 0.2f * e;
      mx = fmaxf(mx, e);
    }
  }
  sred[r][t8] = mx; __syncthreads();
  if (t8 == 0) {
    float v = sred[r][0];
#pragma unroll
    for (int i = 1; i < 8; ++i) v = fmaxf(v, sred[r][i]);
    srow[r] = v;
  }
  __syncthreads();
  mx = srow[r];

  // pass 2: exp into LDS, accumulate row sum
  float sm = 0.0f;
  for (int m = t8; m < NN; m += 8) {
    float p = 0.0f;
    if (adjr[m] != 0) {
      float e = hlv + hrb[m];
      e = (e >= 0.0f) ? e : 0.2f * e;
      p = expf(e - mx);
    }
    sP[r * NN + m] = p;
    sm += p;
  }
  sred[r][t8] = sm; __syncthreads();
  if (t8 == 0) {
    float v = 0.0f;
#pragma unroll
    for (int i = 0; i < 8; ++i) v += sred[r][i];
    srow[16 + r] = 1.0f / v;
  }
  __syncthreads();

  // pass 3: out(16 x 64) = P(16 x 512) @ Wh_head(512 x 64) via WMMA f32.
  // Each of the 4 waves owns a 16-wide slice of the head dim.
  const int lane = threadIdx.x;
  const int half = lane >> 4;
  const int l    = lane & 15;
  const int c0   = threadIdx.y * 16;
  const float rinv = srow[16 + l];                    // scale for A row l
  const float* WhB = Wh + ((size_t)b * NN) * HIDDIM + h * HDD + c0 + l;

  v8f acc = {};
  for (int k = 0; k < NN; k += 4) {
    const int kk = k + half * 2;
    v2f a; a.x = sP[l * NN + kk] * rinv;
           a.y = sP[l * NN + kk + 1] * rinv;
    v2f bb; bb.x = WhB[(size_t)kk * HIDDIM];
            bb.y = WhB[(size_t)(kk + 1) * HIDDIM];
    acc = wmma4(a, bb, acc);
  }
#pragma unroll
  for (int rr = 0; rr < 8; ++rr) {
    const int row = nt * 16 + half * 8 + rr;
    out[((size_t)b * NN + row) * HIDDIM + h * HDD + c0 + l] = acc[rr];
  }
}

// ---------------------------------------------------------------------------
// Fused GRU cell: six WMMA accumulators (i_r,i_z,i_n,h_r,h_z,h_n tiles),
// gate math in epilogue, h_temp written directly.  block=(32,4),
// grid = (HID/64, ROWS/16)
// ---------------------------------------------------------------------------
__global__ void gru_kernel(const float* __restrict__ hattn,
                           const float* __restrict__ hidden,
                           const float* __restrict__ wih,
                           const float* __restrict__ bih,
                           const float* __restrict__ whh,
                           const float* __restrict__ bhh,
                           float* __restrict__ h_temp) {
  const int lane = threadIdx.x, wave = threadIdx.y;
  const int m0 = blockIdx.y * 16;
  const int c0 = blockIdx.x * 64 + wave * 16;   // output col in [0, HID)
  const int half = lane >> 4;
  const int l    = lane & 15;

  const float* Ap = hattn  + (size_t)(m0 + l) * HIDDIM + half * 2;
  const float* Hp = hidden + (size_t)(m0 + l) * HIDDIM + half * 2;
  const float* wi_r = wih + (size_t)(0 * HIDDIM + c0 + l) * HIDDIM + half * 2;
  const float* wi_z = wih + (size_t)(1 * HIDDIM + c0 + l) * HIDDIM + half * 2;
  const float* wi_n = wih + (size_t)(2 * HIDDIM + c0 + l) * HIDDIM + half * 2;
  const float* wh_r = whh + (size_t)(0 * HIDDIM + c0 + l) * HIDDIM + half * 2;
  const float* wh_z = whh + (size_t)(1 * HIDDIM + c0 + l) * HIDDIM + half * 2;
  const float* wh_n = whh + (size_t)(2 * HIDDIM + c0 + l) * HIDDIM + half * 2;

  v8f a_ir = {}, a_iz = {}, a_in = {}, a_hr = {}, a_hz = {}, a_hn = {};
  for (int k = 0; k < HIDDIM; k += 4) {
    v2f a  = *(const v2f*)(Ap + k);
    v2f hh = *(const v2f*)(Hp + k);
    a_ir = wmma4(a,  *(const v2f*)(wi_r + k), a_ir);
    a_iz = wmma4(a,  *(const v2f*)(wi_z + k), a_iz);
    a_in = wmma4(a,  *(const v2f*)(wi_n + k), a_in);
    a_hr = wmma4(hh, *(const v2f*)(wh_r + k), a_hr);
    a_hz = wmma4(hh, *(const v2f*)(wh_z + k), a_hz);
    a_hn = wmma4(hh, *(const v2f*)(wh_n + k), a_hn);
  }

  const int col = c0 + l;
  const float b_ir = bih[col], b_iz = bih[HIDDIM + col], b_in = bih[2 * HIDDIM + col];
  const float b_hr = bhh[col], b_hz = bhh[HIDDIM + col], b_hn = bhh[2 * HIDDIM + col];
#pragma unroll
  for (int rr = 0; rr < 8; ++rr) {
    const int row = m0 + half * 8 + rr;
    const float ir = a_ir[rr] + b_ir, iz = a_iz[rr] + b_iz, in_ = a_in[rr] + b_in;
    const float hrv = a_hr[rr] + b_hr, hz = a_hz[rr] + b_hz, hn = a_hn[rr] + b_hn;
    const float rg = 1.0f / (1.0f + expf(-(ir + hrv)));
    const float zg = 1.0f / (1.0f + expf(-(iz + hz)));
    const float ng = tanhf(in_ + rg * hn);
    const float hp = hidden[(size_t)row * HIDDIM + col];
    h_temp[(size_t)row * HIDDIM + col] = (1.0f - zg) * ng + zg * hp;
  }
}

// ---------------------------------------------------------------------------
extern "C" void kernel_launch(void* const* d_in, const int* in_sizes, int n_in,
                              void* d_out, int out_size, void* d_ws, size_t ws_size,
                              hipStream_t stream) {
  const float* x        = (const float*)d_in[0];
  const int*   adj      = (const int*)  d_in[1];
  const float* hidden   = (const float*)d_in[2];
  const float* enc_w1   = (const float*)d_in[3];
  const float* enc_b1   = (const float*)d_in[4];
  const float* ln1_g    = (const float*)d_in[5];
  const float* ln1_b    = (const float*)d_in[6];
  const float* enc_w2   = (const float*)d_in[7];
  const float* enc_b2   = (const float*)d_in[8];
  const float* posemb   = (const float*)d_in[9];
  const float* gat_w    = (const float*)d_in[10];
  const float* gat_b    = (const float*)d_in[11];
  const float* gat_a    = (const float*)d_in[12];
  const float* gat_ow   = (const float*)d_in[13];
  const float* gat_ob   = (const float*)d_in[14];
  const float* skip_w   = (const float*)d_in[15];
  const float* skip_b   = (const float*)d_in[16];
  const float* gru_wih  = (const float*)d_in[17];
  const float* gru_bih  = (const float*)d_in[18];
  const float* gru_whh  = (const float*)d_in[19];
  const float* gru_bhh  = (const float*)d_in[20];
  const float* dec_w1   = (const float*)d_in[21];
  const float* dec_b1   = (const float*)d_in[22];
  const float* ln2_g    = (const float*)d_in[23];
  const float* ln2_b    = (const float*)d_in[24];
  const float* dec_w2   = (const float*)d_in[25];
  const float* dec_b2   = (const float*)d_in[26];

  float* out    = (float*)d_out;                   // q_values then h_temp
  float* h_temp = out + (size_t)ROWS * ACTD;

  float* ws    = (float*)d_ws;
  float* h1    = ws;                               // 16384 x 512
  float* hsp   = h1    + (size_t)ROWS * 512;       // 16384 x 256
  float* Whb   = hsp   + (size_t)ROWS * HIDDIM;    // 16384 x 256
  float* attn  = Whb   + (size_t)ROWS * HIDDIM;    // 16384 x 256
  float* hattn = attn  + (size_t)ROWS * HIDDIM;    // 16384 x 256
  float* hlb   = hattn + (size_t)ROWS * HIDDIM;    // 32*4*512
  float* hrb   = hlb   + (size_t)NB * NHEADS * NN;
  float* q     = h1;                               // reuse enc1 buffer

  const dim3 blk(32, 4);

  // 1) encoder L1: h1 = x @ enc_w1^T + b1                    (K=128, N=512)
  gemm_wmma_kernel<128><<<dim3(4, ROWS / 16), blk, 0, stream>>>(
      x, enc_w1, enc_b1, nullptr, nullptr, h1, 512);
  // 2) LN + GELU
  ln_gelu_kernel<<<ROWS, 256, 0, stream>>>(h1, ln1_g, ln1_b, 512);
  // 3) encoder L2 + pos_embed: hsp = h1 @ enc_w2^T + b2 + pos (K=512, N=256)
  gemm_wmma_kernel<512><<<dim3(2, ROWS / 16), blk, 0, stream>>>(
      h1, enc_w2, enc_b2, posemb, nullptr, hsp, HIDDIM);
  // 4) Wh = hsp @ gat_w^T + gat_b                            (K=256, N=256)
  gemm_wmma_kernel<256><<<dim3(2, ROWS / 16), blk, 0, stream>>>(
      hsp, gat_w, gat_b, nullptr, nullptr, Whb, HIDDIM);
  // 5) per-head score projections
  gat_score_kernel<<<(ROWS * NHEADS) / 256, 256, 0, stream>>>(Whb, gat_a, hlb, hrb);
  // 6) fused masked softmax + P @ Wh
  gat_attn_kernel<<<dim3(NN / 16, NHEADS, NB), blk, 0, stream>>>(
      Whb, hlb, hrb, adj, attn);
  // 7) out proj, then 8) skip proj accumulated on top        (K=256, N=256)
  gemm_wmma_kernel<256><<<dim3(2, ROWS / 16), blk, 0, stream>>>(
      attn, gat_ow, gat_ob, nullptr, nullptr, hattn, HIDDIM);
  gemm_wmma_kernel<256><<<dim3(2, ROWS / 16), blk, 0, stream>>>(
      hsp, skip_w, skip_b, nullptr, hattn, hattn, HIDDIM);
  // 9) fused GRU -> h_temp (second output)
  gru_kernel<<<dim3(HIDDIM / 64, ROWS / 16), blk, 0, stream>>>(
      hattn, hidden, gru_wih, gru_bih, gru_whh, gru_bhh, h_temp);
  // 10) decoder L1                                           (K=256, N=256)
  gemm_wmma_kernel<256><<<dim3(2, ROWS / 16), blk, 0, stream>>>(
      h_temp, dec_w1, dec_b1, nullptr, nullptr, q, HIDDIM);
  // 11) LN + GELU
  ln_gelu_kernel<<<ROWS, 256, 0, stream>>>(q, ln2_g, ln2_b, HIDDIM);
  // 12) decoder L2 -> q_values (first output)                (K=256, N=32)
  gemm_wmma_kernel<256><<<dim3(1, ROWS / 16), blk, 0, stream>>>(
      q, dec_w2, dec_b2, nullptr, nullptr, out, ACTD);
  (void)in_sizes; (void)n_in; (void)out_size; (void)ws_size;
}